// SlotAttention_72318659330529
// MI455X (gfx1250) — compile-verified
//
#include <hip/hip_runtime.h>
#include <math.h>
#include <stdint.h>

#define B_ 128
#define N_ 8192
#define S_ 4
#define D_ 32
#define ITERS_ 3
#define LN_EPS 1e-5f
#define CCHUNK 16                      // N split into 16 chunks -> 2048 blocks
#define CHUNK_ROWS (N_ / CCHUNK)       // 512 rows per block
#define WAVES 4
#define TPB 128
#define TILES_PER_WAVE (CHUNK_ROWS / (32 * WAVES))  // 4 tiles of 32 rows
#define PART_STRIDE 168                // 4*M + 4*L + 4*32 acc + 32 accX

typedef float v8f __attribute__((ext_vector_type(8)));
typedef float v2f __attribute__((ext_vector_type(2)));
typedef unsigned int u32x4 __attribute__((ext_vector_type(4)));
typedef int i32x4 __attribute__((ext_vector_type(4)));
typedef int i32x8 __attribute__((ext_vector_type(8)));

#if defined(__has_builtin)
#if __has_builtin(__builtin_amdgcn_tensor_load_to_lds)
#define HAVE_TDM 1
#endif
#endif
#ifndef HAVE_TDM
#define HAVE_TDM 0
#endif

__device__ __forceinline__ float sigmoidf_(float x) { return 1.0f / (1.0f + __expf(-x)); }

#if HAVE_TDM
// ---------------------------------------------------------------------------
// Issue one TDM load of a 32x32 f32 tile (row stride 32 elems in memory) into
// LDS at lds_addr, with 1 DWORD of LDS padding after every 32 DWORDs so the
// LDS image has row stride 33 floats (bank-conflict-free row access).
// D# fields per CDNA5 ISA 8.3/8.4.  All descriptor words made wave-uniform.
// ---------------------------------------------------------------------------
__device__ __forceinline__ void tdm_load_tile32(const float* gsrc, uint32_t lds_addr) {
    uint64_t ga    = (uint64_t)(uintptr_t)gsrc;
    uint32_t ga_lo = __builtin_amdgcn_readfirstlane((uint32_t)ga);
    uint32_t ga_hi = __builtin_amdgcn_readfirstlane((uint32_t)(ga >> 32));
    uint32_t la    = __builtin_amdgcn_readfirstlane(lds_addr);

    u32x4 g0 = { 1u,                                    // count=1 (valid user D#)
                 la,                                    // lds_addr [63:32]
                 ga_lo,                                 // global_addr [95:64]
                 (ga_hi & 0x01FFFFFFu) | (2u << 30) };  // ga[56:32] | type=2

    // word0: data_size=2 (4B) | pad_enable | pad_interval=4 (32 dwords) | pad_amount=0 (1 dword)
    i32x8 g1 = { (int)((2u << 16) | (1u << 20) | (4u << 22)),
                 (int)(32u << 16),    // tensor_dim0 = 32 (bits 79:48)
                 (int)(32u << 16),    // tensor_dim1 = 32 (bits 111:80)
                 (int)(32u << 16),    // tile_dim0   = 32 (bits 127:112)
                 32,                  // tile_dim1   = 32 (bits 143:128)
                 32,                  // tensor_dim0_stride = 32 (bits 207:160)
                 0, 0 };
    i32x4 z4 = { 0, 0, 0, 0 };
#if __clang_major__ >= 23
    i32x8 z8 = { 0, 0, 0, 0, 0, 0, 0, 0 };
    __builtin_amdgcn_tensor_load_to_lds(g0, g1, z4, z4, z8, 0);
#else
    __builtin_amdgcn_tensor_load_to_lds(g0, g1, z4, z4, 0);
#endif
}
#endif

// ---------------------------------------------------------------------------
// slots = mu + noise * exp(logsigma)
// ---------------------------------------------------------------------------
__global__ void k_init(const float* __restrict__ mu, const float* __restrict__ lsig,
                       const float* __restrict__ noise, float* __restrict__ slots) {
    int i = blockIdx.x * blockDim.x + threadIdx.x;
    if (i < B_ * S_ * D_) {
        int d = i & (D_ - 1);
        slots[i] = mu[d] + noise[i] * __expf(lsig[d]);
    }
}

// ---------------------------------------------------------------------------
// Per-batch: q = LN(slots) @ Wq^T ; qk = scale * (q @ Wk)  -> [S][D]
// ---------------------------------------------------------------------------
__global__ void k_qproj(const float* __restrict__ slots, const float* __restrict__ Wq,
                        const float* __restrict__ Wk, const float* __restrict__ lns_g,
                        const float* __restrict__ lns_b, float* __restrict__ qk) {
    const int b = blockIdx.x, tid = threadIdx.x;
    const int s = tid >> 5, d = tid & 31;
    __shared__ float s_s[S_][D_], s_sn[S_][D_], s_q[S_][D_], s_m[S_], s_r[S_];
    s_s[s][d] = slots[b * 128 + tid];
    __syncthreads();
    if (tid < S_) {
        float sm = 0.f, sq = 0.f;
        for (int j = 0; j < D_; j++) { float t = s_s[tid][j]; sm += t; sq += t * t; }
        sm *= (1.0f / D_);
        float var = sq * (1.0f / D_) - sm * sm;
        s_m[tid] = sm;
        s_r[tid] = rsqrtf(var + LN_EPS);
    }
    __syncthreads();
    s_sn[s][d] = (s_s[s][d] - s_m[s]) * s_r[s] * lns_g[d] + lns_b[d];
    __syncthreads();
    {
        float acc = 0.f;
        for (int j = 0; j < D_; j++) acc += s_sn[s][j] * Wq[d * D_ + j];
        s_q[s][d] = acc;
    }
    __syncthreads();
    {
        const float scale = 0.17677669529663687f;  // 1/sqrt(32)
        float acc = 0.f;
        for (int j = 0; j < D_; j++) acc += s_q[s][j] * Wk[j * D_ + d];
        qk[b * 128 + tid] = acc * scale;
    }
}

// ---------------------------------------------------------------------------
// Heavy pass: TDM double-buffered x tiles -> LDS, row LN, logits via chained
// V_WMMA_F32_16X16X4_F32, flash-style online softmax,
// acc[s] += p * xn, accX += xn.  One partial record per (b, chunk).
// ---------------------------------------------------------------------------
__global__ __launch_bounds__(TPB)
void k_attn(const float* __restrict__ x, const float* __restrict__ lnx_g,
            const float* __restrict__ lnx_b, const float* __restrict__ qk,
            float* __restrict__ part) {
    const int chunk = blockIdx.x, b = blockIdx.y;
    const int tid = threadIdx.x, w = tid >> 5, lane = tid & 31;

    __shared__ float s_qk[S_][D_];
    __shared__ float s_g[D_], s_bb[D_];
    __shared__ float s_xn[WAVES][2][32][33];   // double-buffered, padded rows
    __shared__ float s_w[WAVES][32][S_];       // logits -> exp weights
    __shared__ float s_sc[WAVES][S_];          // rescale broadcast / merge factors
    __shared__ float s_red[WAVES][PART_STRIDE];

    s_qk[tid >> 5][tid & 31] = qk[b * 128 + tid];
    if (tid < 32) { s_g[tid] = lnx_g[tid]; s_bb[tid] = lnx_b[tid]; }
    __syncthreads();

    // B fragment for V_WMMA_F32_16X16X4_F32:  B[k][j] = qk[j][k], cols 4..15 = 0.
    // Layout (wave32): lanes 0-15: VGPR0=K0,VGPR1=K1 ; lanes 16-31: VGPR0=K2,VGPR1=K3.
    const int col = lane & 15, hk = lane >> 4;
    v2f bfrag[8];
#pragma unroll
    for (int kk = 0; kk < 8; kk++) {
        int k0 = 4 * kk + 2 * hk;
        bfrag[kk].x = (col < S_) ? s_qk[col][k0]     : 0.0f;
        bfrag[kk].y = (col < S_) ? s_qk[col][k0 + 1] : 0.0f;
    }

    float Ms = -3.0e38f, Ls = 0.0f;                  // lanes 0..3: slot state
    float a0 = 0.f, a1 = 0.f, a2 = 0.f, a3 = 0.f, sx = 0.f;

    const float* xbase = x + ((size_t)b * N_ + (size_t)chunk * CHUNK_ROWS) * D_;
#if HAVE_TDM
    const uint32_t lds0 = (uint32_t)(uintptr_t)&s_xn[w][0][0][0];
    const uint32_t lds1 = (uint32_t)(uintptr_t)&s_xn[w][1][0][0];
    tdm_load_tile32(xbase + (size_t)((0 * WAVES + w) * 32) * D_, lds0);
#endif

    for (int t = 0; t < TILES_PER_WAVE; t++) {
        const int buf = t & 1;
#if HAVE_TDM
        if (t + 1 < TILES_PER_WAVE) {
            tdm_load_tile32(xbase + (size_t)(((t + 1) * WAVES + w) * 32) * D_,
                            (t & 1) ? lds0 : lds1);
            __builtin_amdgcn_s_wait_tensorcnt((short)1);   // tile t done
        } else {
            __builtin_amdgcn_s_wait_tensorcnt((short)0);
        }
        __builtin_amdgcn_wave_barrier();
#else
        {   // fallback: coalesced float4 loads -> LDS
            const int row0 = (t * WAVES + w) * 32;
            const float4* xp = (const float4*)(xbase + (size_t)row0 * D_);
#pragma unroll
            for (int rr = 0; rr < 8; rr++) {
                int r = rr * 4 + (lane >> 3);
                float4 vx = xp[r * 8 + (lane & 7)];
                int cc = (lane & 7) * 4;
                s_xn[w][buf][r][cc + 0] = vx.x; s_xn[w][buf][r][cc + 1] = vx.y;
                s_xn[w][buf][r][cc + 2] = vx.z; s_xn[w][buf][r][cc + 3] = vx.w;
            }
            __builtin_amdgcn_wave_barrier();
        }
#endif

        // ---- LayerNorm: lane owns row 'lane' ----
        {
            float sm = 0.f, sq = 0.f;
            for (int d = 0; d < D_; d++) { float tv = s_xn[w][buf][lane][d]; sm += tv; sq += tv * tv; }
            sm *= (1.0f / D_);
            float var  = sq * (1.0f / D_) - sm * sm;
            float rstd = rsqrtf(var + LN_EPS);
            for (int d = 0; d < D_; d++)
                s_xn[w][buf][lane][d] = (s_xn[w][buf][lane][d] - sm) * rstd * s_g[d] + s_bb[d];
        }
        __builtin_amdgcn_wave_barrier();

        // ---- logits = xn_tile @ qk^T via chained f32 WMMA (two 16-row subtiles) ----
#pragma unroll
        for (int sub = 0; sub < 2; sub++) {
            v8f cacc = {0.f, 0.f, 0.f, 0.f, 0.f, 0.f, 0.f, 0.f};
            const int arow = (lane & 15) + 16 * sub;
#pragma unroll
            for (int kk = 0; kk < 8; kk++) {
                v2f af;
                int k0 = 4 * kk + 2 * hk;
                af.x = s_xn[w][buf][arow][k0];
                af.y = s_xn[w][buf][arow][k0 + 1];
                cacc = __builtin_amdgcn_wmma_f32_16x16x4_f32(
                    false, af, false, bfrag[kk], (short)0, cacc, false, false);
            }
            // C/D layout: VGPR r -> M=r (lanes 0-15) / M=r+8 (lanes 16-31), N=lane&15
            if (col < S_) {
#pragma unroll
                for (int r = 0; r < 8; r++) {
                    int mrow = r + 8 * hk;
                    s_w[w][sub * 16 + mrow][col] = cacc[r];
                }
            }
        }
        __builtin_amdgcn_wave_barrier();

        // ---- online softmax update: lanes 0..3 (slot = lane) ----
        if (lane < S_) {
            float tm = -3.0e38f;
            for (int r = 0; r < 32; r++) tm = fmaxf(tm, s_w[w][r][lane]);
            float Mn = fmaxf(Ms, tm);
            float sc = __expf(Ms - Mn);
            float ssum = 0.f;
            for (int r = 0; r < 32; r++) {
                float e = __expf(s_w[w][r][lane] - Mn);
                s_w[w][r][lane] = e;
                ssum += e;
            }
            Ls = Ls * sc + ssum;
            Ms = Mn;
            s_sc[w][lane] = sc;
        }
        __builtin_amdgcn_wave_barrier();

        // ---- acc rescale + accumulate (lane owns dim d=lane) ----
        {
            float c0 = s_sc[w][0], c1 = s_sc[w][1], c2 = s_sc[w][2], c3 = s_sc[w][3];
            a0 *= c0; a1 *= c1; a2 *= c2; a3 *= c3;
            for (int r = 0; r < 32; r++) {
                float xv = s_xn[w][buf][r][lane];
                sx += xv;
                a0 += s_w[w][r][0] * xv;
                a1 += s_w[w][r][1] * xv;
                a2 += s_w[w][r][2] * xv;
                a3 += s_w[w][r][3] * xv;
            }
        }
    }

    // ---- per-wave partial -> LDS, block merge, write one partial record ----
    if (lane < S_) { s_red[w][lane] = Ms; s_red[w][4 + lane] = Ls; }
    s_red[w][8 + 0 * 32 + lane] = a0;
    s_red[w][8 + 1 * 32 + lane] = a1;
    s_red[w][8 + 2 * 32 + lane] = a2;
    s_red[w][8 + 3 * 32 + lane] = a3;
    s_red[w][136 + lane] = sx;
    __syncthreads();

    float* po = part + (size_t)(b * CCHUNK + chunk) * PART_STRIDE;
    if (tid < S_) {
        float Mg = -3.0e38f;
        for (int ww = 0; ww < WAVES; ww++) Mg = fmaxf(Mg, s_red[ww][tid]);
        float Lg = 0.f;
        for (int ww = 0; ww < WAVES; ww++) {
            float f = __expf(s_red[ww][tid] - Mg);
            s_sc[ww][tid] = f;
            Lg += s_red[ww][4 + tid] * f;
        }
        po[tid] = Mg;
        po[4 + tid] = Lg;
    }
    __syncthreads();
    {
        int s = tid >> 5, d = tid & 31;
        float acc = 0.f;
        for (int ww = 0; ww < WAVES; ww++) acc += s_red[ww][8 + s * 32 + d] * s_sc[ww][s];
        po[8 + tid] = acc;
    }
    if (tid < 32) {
        float ax = 0.f;
        for (int ww = 0; ww < WAVES; ww++) ax += s_red[ww][136 + tid];
        po[136 + tid] = ax;
    }
}

// ---------------------------------------------------------------------------
// Merge partials, upd = ((acc/L + eps*accX)/(1+N*eps)) @ Wv^T, GRU, MLP.
// ---------------------------------------------------------------------------
__global__ void k_update(const float* __restrict__ part, float* __restrict__ slots,
                         const float* __restrict__ Wv, const float* __restrict__ Wih,
                         const float* __restrict__ Whh, const float* __restrict__ bih,
                         const float* __restrict__ bhh, const float* __restrict__ W1,
                         const float* __restrict__ b1, const float* __restrict__ W2,
                         const float* __restrict__ b2, const float* __restrict__ lnff_g,
                         const float* __restrict__ lnff_b, float* __restrict__ out,
                         int write_out) {
    const int b = blockIdx.x, tid = threadIdx.x;
    const int s = tid >> 5, d = tid & 31;
    __shared__ float s_slot[S_][D_], s_a[S_][D_], s_upd[S_][D_];
    __shared__ float s_h[S_][D_], s_xn[S_][D_], s_hh[S_][D_];
    __shared__ float s_gi[S_][96], s_gh[S_][96];
    __shared__ float s_f[CCHUNK][S_], s_L[S_], s_stat[S_][2];

    s_slot[s][d] = slots[b * 128 + tid];
    const float* pb = part + (size_t)b * CCHUNK * PART_STRIDE;
    if (tid < S_) {
        float Mg = -3.0e38f;
        for (int cg = 0; cg < CCHUNK; cg++) Mg = fmaxf(Mg, pb[cg * PART_STRIDE + tid]);
        float Lg = 0.f;
        for (int cg = 0; cg < CCHUNK; cg++) {
            float f = __expf(pb[cg * PART_STRIDE + tid] - Mg);
            s_f[cg][tid] = f;
            Lg += pb[cg * PART_STRIDE + 4 + tid] * f;
        }
        s_L[tid] = Lg;
    }
    __syncthreads();
    {
        float acc = 0.f, ax = 0.f;
        for (int cg = 0; cg < CCHUNK; cg++) {
            acc += pb[cg * PART_STRIDE + 8 + s * 32 + d] * s_f[cg][s];
            ax  += pb[cg * PART_STRIDE + 136 + d];
        }
        const float inv = 1.0f / (1.0f + (float)N_ * 1e-8f);
        s_a[s][d] = (acc / s_L[s] + 1e-8f * ax) * inv;
    }
    __syncthreads();
    {   // upd[s][d] = sum_j a[s][j] * Wv[d][j]
        float u = 0.f;
        for (int j = 0; j < D_; j++) u += s_a[s][j] * Wv[d * D_ + j];
        s_upd[s][d] = u;
    }
    __syncthreads();
    for (int idx = tid; idx < S_ * 96; idx += 128) {
        int ss = idx / 96, g = idx % 96;
        float gi = bih[g], gh = bhh[g];
        for (int j = 0; j < D_; j++) {
            gi += s_upd[ss][j] * Wih[g * D_ + j];
            gh += s_slot[ss][j] * Whh[g * D_ + j];
        }
        s_gi[ss][g] = gi;
        s_gh[ss][g] = gh;
    }
    __syncthreads();
    {
        float r = sigmoidf_(s_gi[s][d] + s_gh[s][d]);
        float z = sigmoidf_(s_gi[s][32 + d] + s_gh[s][32 + d]);
        float n = tanhf(s_gi[s][64 + d] + r * s_gh[s][64 + d]);
        s_h[s][d] = (1.0f - z) * n + z * s_slot[s][d];
    }
    __syncthreads();
    if (tid < S_) {
        float sm = 0.f, sq = 0.f;
        for (int j = 0; j < D_; j++) { float t = s_h[tid][j]; sm += t; sq += t * t; }
        sm *= (1.0f / D_);
        float var = sq * (1.0f / D_) - sm * sm;
        s_stat[tid][0] = sm;
        s_stat[tid][1] = rsqrtf(var + LN_EPS);
    }
    __syncthreads();
    s_xn[s][d] = (s_h[s][d] - s_stat[s][0]) * s_stat[s][1] * lnff_g[d] + lnff_b[d];
    __syncthreads();
    {
        float hh = b1[d];
        for (int j = 0; j < D_; j++) hh += s_xn[s][j] * W1[d * D_ + j];
        s_hh[s][d] = fmaxf(hh, 0.0f);
    }
    __syncthreads();
    {
        float o = s_h[s][d] + b2[d];
        for (int j = 0; j < D_; j++) o += s_hh[s][j] * W2[d * D_ + j];
        slots[b * 128 + tid] = o;
        if (write_out) out[b * 128 + tid] = o;
    }
}

// ---------------------------------------------------------------------------
extern "C" void kernel_launch(void* const* d_in, const int* in_sizes, int n_in,
                              void* d_out, int out_size, void* d_ws, size_t ws_size,
                              hipStream_t stream) {
    const float* x      = (const float*)d_in[0];
    const float* noise  = (const float*)d_in[1];
    const float* mu     = (const float*)d_in[2];
    const float* lsig   = (const float*)d_in[3];
    const float* Wq     = (const float*)d_in[4];
    const float* Wk     = (const float*)d_in[5];
    const float* Wv     = (const float*)d_in[6];
    const float* Wih    = (const float*)d_in[7];
    const float* Whh    = (const float*)d_in[8];
    const float* bih    = (const float*)d_in[9];
    const float* bhh    = (const float*)d_in[10];
    const float* W1     = (const float*)d_in[11];
    const float* b1     = (const float*)d_in[12];
    const float* W2     = (const float*)d_in[13];
    const float* b2     = (const float*)d_in[14];
    const float* lnx_g  = (const float*)d_in[15];
    const float* lnx_b  = (const float*)d_in[16];
    const float* lns_g  = (const float*)d_in[17];
    const float* lns_b  = (const float*)d_in[18];
    const float* lnff_g = (const float*)d_in[19];
    const float* lnff_b = (const float*)d_in[20];

    float* ws_slots = (float*)d_ws;                  // B*S*D
    float* ws_qk    = ws_slots + B_ * S_ * D_;       // B*S*D
    float* ws_part  = ws_qk + B_ * S_ * D_;          // B*CCHUNK*PART_STRIDE

    k_init<<<(B_ * S_ * D_ + 255) / 256, 256, 0, stream>>>(mu, lsig, noise, ws_slots);

    for (int it = 0; it < ITERS_; ++it) {
        k_qproj<<<B_, 128, 0, stream>>>(ws_slots, Wq, Wk, lns_g, lns_b, ws_qk);
        dim3 grid(CCHUNK, B_);
        k_attn<<<grid, TPB, 0, stream>>>(x, lnx_g, lnx_b, ws_qk, ws_part);
        k_update<<<B_, 128, 0, stream>>>(ws_part, ws_slots, Wv, Wih, Whh, bih, bhh,
                                         W1, b1, W2, b2, lnff_g, lnff_b,
                                         (float*)d_out, (it == ITERS_ - 1) ? 1 : 0);
    }
}